// MyGAT_29386166239368
// MI455X (gfx1250) — compile-verified
//
#include <hip/hip_runtime.h>
#include <hip/hip_bf16.h>
#include <math.h>

typedef __attribute__((ext_vector_type(2))) float v2f;
typedef __attribute__((ext_vector_type(8))) float v8f;

#define NEG_SLOPE 0.2f
#define EPS_DENOM 1e-16f

// LDS row stride for the staged A tile: 132 % 64 == 4  => lanes reading rows
// at stride 132 floats hit banks {0,4,...,60}: conflict-free b64 reads, and
// 132 % 4 == 0 keeps float4 staging stores 16B-aligned.
#define ASTRIDE 132

// ---------------------------------------------------------------------------
// Helpers
// ---------------------------------------------------------------------------
__device__ __forceinline__ void atomicMaxF32(float* addr, float val) {
    // Sign-aware bit-punned float max: works with -inf init (0xFF800000).
    if (val >= 0.0f) {
        atomicMax((int*)addr, __float_as_int(val));
    } else {
        atomicMin((unsigned int*)addr, __float_as_uint(val));
    }
}

__global__ void fill_kernel(float* __restrict__ p, float v, long long n) {
    long long i = (long long)blockIdx.x * blockDim.x + threadIdx.x;
    if (i < n) p[i] = v;
}

// ---------------------------------------------------------------------------
// GEMM: Hout[n, outC] = act(A[n, 0:128]) @ W[128, outC]
// act = (APPLY ? relu(x + bias[k]) : x)  -- prev layer's epilogue fused into
// the A-tile staging.  Block = 256 threads = 8 waves; block stages its A rows
// (8/COLTILES row-tiles of 16) into LDS once, then each wave computes one
// 16x16 tile with fp32 WMMA (V_WMMA_F32_16X16X4_F32), K=128 in 32 steps.
// ---------------------------------------------------------------------------
template<int COLTILES, bool APPLY>
__global__ void gat_gemm_wmma(const float* __restrict__ A,
                              const float* __restrict__ W,
                              const float* __restrict__ bias,
                              float* __restrict__ Hout,
                              int nRows) {
    constexpr int outC        = COLTILES * 16;
    constexpr int ROWTILES_PB = 8 / COLTILES;       // row tiles per block
    constexpr int ROWS        = ROWTILES_PB * 16;   // A rows staged per block

    __shared__ float sA[ROWS * ASTRIDE];

    const int tid  = threadIdx.x;
    const int lane = tid & 31;
    const int wid  = tid >> 5;

    const int rowTiles     = (nRows + 15) >> 4;
    const int blockRowBase = blockIdx.x * ROWS;

    // ---- cooperative stage: A tile -> LDS, fusing bias+ReLU once ----------
    const int totalVec4 = ROWS * 32;                // 128 floats/row = 32x f4
    for (int i = tid; i < totalVec4; i += 256) {
        const int r  = i >> 5;
        const int c4 = (i & 31) << 2;
        int gRow = blockRowBase + r;
        if (gRow >= nRows) gRow = nRows - 1;        // clamp (harmless dup)
        float4 v = *(const float4*)(A + (size_t)gRow * 128 + c4);
        if (APPLY) {
            v.x += bias[c4 + 0]; v.y += bias[c4 + 1];
            v.z += bias[c4 + 2]; v.w += bias[c4 + 3];
            v.x = v.x > 0.f ? v.x : 0.f;  v.y = v.y > 0.f ? v.y : 0.f;
            v.z = v.z > 0.f ? v.z : 0.f;  v.w = v.w > 0.f ? v.w : 0.f;
        }
        *(float4*)(&sA[r * ASTRIDE + c4]) = v;
    }
    __syncthreads();

    // ---- per-wave 16x16 tile ----------------------------------------------
    const int rowTile = blockIdx.x * ROWTILES_PB + (wid / COLTILES);
    const int colTile = wid % COLTILES;
    const bool active = rowTile < rowTiles;         // wave-uniform
    if (!active) return;

    const int rowBase  = rowTile << 4;
    const int colBase  = colTile << 4;
    const int half     = lane >> 4;                 // 0: K {0,1}; 1: K {2,3}
    const int l15      = lane & 15;
    const int localRow = (wid / COLTILES) * 16 + l15;
    const int kk       = half << 1;

    const float* __restrict__ sRow = &sA[localRow * ASTRIDE];

    v8f acc = {0.f, 0.f, 0.f, 0.f, 0.f, 0.f, 0.f, 0.f};

    #pragma unroll 8
    for (int k0 = 0; k0 < 128; k0 += 4) {
        const int ka = k0 + kk;
        const float2 av = *(const float2*)(sRow + ka);   // ds_load_b64, no conflicts
        v2f a = {av.x, av.y};
        v2f b = { W[(size_t)ka * outC + colBase + l15],
                  W[(size_t)(ka + 1) * outC + colBase + l15] };
        acc = __builtin_amdgcn_wmma_f32_16x16x4_f32(
                  /*neg_a=*/false, a, /*neg_b=*/false, b,
                  /*c_mod=*/(short)0, acc, /*reuse_a=*/false, /*reuse_b=*/false);
    }

    #pragma unroll
    for (int j = 0; j < 8; ++j) {
        const int row = rowBase + j + (half << 3);
        if (row < nRows)
            Hout[(size_t)row * outC + colBase + l15] = acc[j];
    }
}

// ---------------------------------------------------------------------------
// Per-node attention coefficients: as_[n,h] = <h[n,h,:], a_src[h,:]>, same dst
// ---------------------------------------------------------------------------
__global__ void gat_alpha(const float* __restrict__ Hf,
                          const float* __restrict__ a_src,
                          const float* __restrict__ a_dst,
                          float* __restrict__ as_, float* __restrict__ ad_,
                          int nNodes, int Hn) {
    int t = blockIdx.x * blockDim.x + threadIdx.x;
    if (t >= nNodes * Hn) return;
    const int n = t / Hn;
    const int h = t - n * Hn;
    const float* __restrict__ hp = Hf + (size_t)n * Hn * 32 + h * 32;
    const float* __restrict__ sp = a_src + h * 32;
    const float* __restrict__ dp = a_dst + h * 32;
    float s = 0.f, d = 0.f;
    #pragma unroll 8
    for (int c = 0; c < 32; ++c) {
        const float v = hp[c];
        s = fmaf(v, sp[c], s);
        d = fmaf(v, dp[c], d);
    }
    as_[t] = s;
    ad_[t] = d;
}

// ---------------------------------------------------------------------------
// Edge pass 1: segment max of leaky(alpha_s[src]+alpha_d[dst]) per (dst, h)
// ---------------------------------------------------------------------------
__global__ void gat_edge_max(const long long* __restrict__ ei,
                             long long E, long long Etot,
                             const float* __restrict__ as_,
                             const float* __restrict__ ad_,
                             float* __restrict__ m_, int Hn) {
    long long t = (long long)blockIdx.x * blockDim.x + threadIdx.x;
    if (t >= Etot * Hn) return;
    const long long e = t / Hn;
    const int h = (int)(t - e * Hn);
    long long s, d;
    if (e < E) { s = ei[e]; d = ei[E + e]; } else { s = d = e - E; }
    float v = as_[s * Hn + h] + ad_[d * Hn + h];
    v = v > 0.f ? v : NEG_SLOPE * v;
    atomicMaxF32(&m_[d * Hn + h], v);
}

// ---------------------------------------------------------------------------
// Edge pass 2: denom[dst,h] += exp(e - m[dst,h])
// ---------------------------------------------------------------------------
__global__ void gat_edge_expsum(const long long* __restrict__ ei,
                                long long E, long long Etot,
                                const float* __restrict__ as_,
                                const float* __restrict__ ad_,
                                const float* __restrict__ m_,
                                float* __restrict__ dn_, int Hn) {
    long long t = (long long)blockIdx.x * blockDim.x + threadIdx.x;
    if (t >= Etot * Hn) return;
    const long long e = t / Hn;
    const int h = (int)(t - e * Hn);
    long long s, d;
    if (e < E) { s = ei[e]; d = ei[E + e]; } else { s = d = e - E; }
    float v = as_[s * Hn + h] + ad_[d * Hn + h];
    v = v > 0.f ? v : NEG_SLOPE * v;
    atomicAdd(&dn_[d * Hn + h], expf(v - m_[d * Hn + h]));
}

// ---------------------------------------------------------------------------
// Edge pass 3: agg[dst,h,:] += alpha * h[src,h,:].  One wave32 per edge;
// each lane owns Hn contiguous channels (Hn=4: one float4 + 4 f32 atomics).
// Per-layer working set (~135MB) fits MI455X's 192MB L2: the random gathers
// and global_atomic_add_f32 stay L2-resident.
// ---------------------------------------------------------------------------
__global__ void gat_edge_aggregate(const long long* __restrict__ ei,
                                   long long E, long long Etot,
                                   const float* __restrict__ as_,
                                   const float* __restrict__ ad_,
                                   const float* __restrict__ m_,
                                   const float* __restrict__ dn_,
                                   const float* __restrict__ Hf,
                                   float* __restrict__ agg, int Hn) {
    const int lane = threadIdx.x & 31;
    const long long wave = (long long)blockIdx.x * (blockDim.x >> 5)
                         + (threadIdx.x >> 5);
    if (wave >= Etot) return;
    long long s, d;
    if (wave < E) { s = ei[wave]; d = ei[E + wave]; } else { s = d = wave - E; }

    const int HC   = Hn << 5;        // Hn*32 channels total
    const int base = lane * Hn;      // this lane's channel chunk
    const int head = base >> 5;      // chunk never crosses a head boundary

    float v = as_[s * Hn + head] + ad_[d * Hn + head];
    v = v > 0.f ? v : NEG_SLOPE * v;
    const float w = expf(v - m_[d * Hn + head])
                  / (dn_[d * Hn + head] + EPS_DENOM);

    const float* __restrict__ hp = Hf + (size_t)s * HC + base;
    float* __restrict__ op       = agg + (size_t)d * HC + base;

    if (Hn == 4) {
        const float4 hv = *(const float4*)hp;
        atomicAdd(op + 0, hv.x * w);
        atomicAdd(op + 1, hv.y * w);
        atomicAdd(op + 2, hv.z * w);
        atomicAdd(op + 3, hv.w * w);
    } else {
        atomicAdd(op, hp[0] * w);
    }
}

// ---------------------------------------------------------------------------
// Final epilogue: out = agg + b  (last layer has no ReLU)
// ---------------------------------------------------------------------------
__global__ void gat_bias_out(const float* __restrict__ agg,
                             const float* __restrict__ b,
                             float* __restrict__ out,
                             long long total, int C) {
    long long i = (long long)blockIdx.x * blockDim.x + threadIdx.x;
    if (i >= total) return;
    out[i] = agg[i] + b[(int)(i % C)];
}

// ---------------------------------------------------------------------------
// Launch
// ---------------------------------------------------------------------------
extern "C" void kernel_launch(void* const* d_in, const int* in_sizes, int n_in,
                              void* d_out, int out_size, void* d_ws, size_t ws_size,
                              hipStream_t stream) {
    (void)n_in; (void)out_size; (void)ws_size;

    const float*     x  = (const float*)d_in[0];
    const long long* ei = (const long long*)d_in[1];     // int64 edge index
    const float* Wm[3]   = {(const float*)d_in[2], (const float*)d_in[6],  (const float*)d_in[10]};
    const float* asrc[3] = {(const float*)d_in[3], (const float*)d_in[7],  (const float*)d_in[11]};
    const float* adst[3] = {(const float*)d_in[4], (const float*)d_in[8],  (const float*)d_in[12]};
    const float* bb[3]   = {(const float*)d_in[5], (const float*)d_in[9],  (const float*)d_in[13]};

    const int       Nn   = in_sizes[0] / 128;            // 100000 nodes
    const long long E    = (long long)in_sizes[1] / 2;   // 1.6M edges
    const long long Etot = E + Nn;                       // + self loops

    // Workspace carve-out (floats)
    float* ws    = (float*)d_ws;
    float* hfeat = ws;                                   // N*128
    float* agg   = hfeat + (size_t)Nn * 128;             // N*128 (also next A)
    float* as_   = agg   + (size_t)Nn * 128;             // N*4
    float* ad_   = as_   + (size_t)Nn * 4;               // N*4
    float* m_    = ad_   + (size_t)Nn * 4;               // N*4
    float* dn_   = m_    + (size_t)Nn * 4;               // N*4

    const int rowTiles = (Nn + 15) / 16;

    for (int l = 0; l < 3; ++l) {
        const int Hn   = (l < 2) ? 4 : 1;
        const int outC = Hn * 32;

        // 1) GEMM (fp32 WMMA, LDS-staged A with fused prev bias+ReLU)
        if (l == 0) {
            gat_gemm_wmma<8, false><<<rowTiles, 256, 0, stream>>>(
                x, Wm[0], bb[0], hfeat, Nn);
        } else if (l == 1) {
            gat_gemm_wmma<8, true><<<rowTiles, 256, 0, stream>>>(
                agg, Wm[1], bb[0], hfeat, Nn);          // reads agg BEFORE re-zero
        } else {
            gat_gemm_wmma<2, true><<<(rowTiles + 3) / 4, 256, 0, stream>>>(
                agg, Wm[2], bb[1], hfeat, Nn);
        }

        // 2) Per-node attention dots
        const int tA = Nn * Hn;
        gat_alpha<<<(tA + 255) / 256, 256, 0, stream>>>(
            hfeat, asrc[l], adst[l], as_, ad_, Nn, Hn);

        // 3) Init segment-reduction buffers (agg re-zeroed AFTER GEMM read it)
        fill_kernel<<<(unsigned)((tA + 255) / 256), 256, 0, stream>>>(
            m_, -3.402823466e38f, (long long)tA);
        fill_kernel<<<(unsigned)((tA + 255) / 256), 256, 0, stream>>>(
            dn_, 0.0f, (long long)tA);
        const long long aggN = (long long)Nn * outC;
        fill_kernel<<<(unsigned)((aggN + 255) / 256), 256, 0, stream>>>(
            agg, 0.0f, aggN);

        // 4) Edge passes (softmax over incoming edges per dst, then aggregate)
        const long long tE = Etot * Hn;
        gat_edge_max<<<(unsigned)((tE + 255) / 256), 256, 0, stream>>>(
            ei, E, Etot, as_, ad_, m_, Hn);
        gat_edge_expsum<<<(unsigned)((tE + 255) / 256), 256, 0, stream>>>(
            ei, E, Etot, as_, ad_, m_, dn_, Hn);
        gat_edge_aggregate<<<(unsigned)((Etot + 7) / 8), 256, 0, stream>>>(
            ei, E, Etot, as_, ad_, m_, dn_, hfeat, agg, Hn);
    }

    // Final bias add -> d_out (no ReLU on last layer)
    const long long totalOut = (long long)Nn * 32;
    gat_bias_out<<<(unsigned)((totalOut + 255) / 256), 256, 0, stream>>>(
        agg, bb[2], (float*)d_out, totalOut, 32);
}